// FlexicubesHelper_72962904424814
// MI455X (gfx1250) — compile-verified
//
#include <hip/hip_runtime.h>
#include <hip/hip_bf16.h>

// ---------------- problem constants ----------------
#define RES   96
#define NV1   97                     // RES+1
#define NVTOT 912673                 // 97^3
#define NCUBE 884736                 // 96^3
#define NEF   866400                 // 96*95*95 per edge family
#define NEDGE 2599200                // 3 * NEF
#define NTRI  5198400                // 2 * NEDGE

#define OFF_VD    0
#define OFF_TRI   2654208            // 3*NCUBE
#define OFF_MASK  18249408           // OFF_TRI + 3*NTRI
#define OFF_REG   23447808           // OFF_MASK + NTRI
#define OFF_EMPTY 23447809
#define OFF_ZS    23447810

#define QEF_REG 0.001f
#define WSCALE  0.99f
#define DSCALE  ((1.0f - 1e-8f) / (2.0f * (float)RES))   // (0.5-(-0.5)-1e-8)/(2*RES)
#define FLT_BIG 3.402823466e+38f

// Reduction-1 config
#define G1 1024
// ws layout (floats): [0..7] scalars, [8..8+5*G1) k1 partials, then k3 partials
#define WS_SCAL   0
#define WS_PART1  8
#define WS_PART3  (8 + 5 * G1)
#define G3        (NCUBE / 256)      // 3456 blocks, exact

// ---------------- CDNA5 async global->LDS path ----------------
#if defined(__HIP_DEVICE_COMPILE__) && __has_builtin(__builtin_amdgcn_global_load_async_to_lds_b32)
#define HAVE_ASYNC_LDS 1
#else
#define HAVE_ASYNC_LDS 0
#endif
#if defined(__HIP_DEVICE_COMPILE__) && __has_builtin(__builtin_amdgcn_s_wait_asynccnt)
#define HAVE_WAIT_ASYNC 1
#else
#define HAVE_WAIT_ASYNC 0
#endif

typedef __attribute__((address_space(1))) int gas_int;
typedef __attribute__((address_space(3))) int las_int;

__device__ __forceinline__ void async_cp_f32(const float* g, float* l) {
#if HAVE_ASYNC_LDS
  __builtin_amdgcn_global_load_async_to_lds_b32(
      (gas_int*)g,
      (las_int*)l,
      0, 0);
#else
  *l = *g;
#endif
}

__device__ __forceinline__ void async_wait_all() {
#if HAVE_ASYNC_LDS
# if HAVE_WAIT_ASYNC
  __builtin_amdgcn_s_wait_asynccnt(0);
# else
  asm volatile("s_wait_asynccnt 0" ::: "memory");
# endif
#endif
}

// sdf with the conditional analytic center/boundary update applied
__device__ __forceinline__ float sdf_at(const float* __restrict__ level,
                                        int i, int j, int k,
                                        float zs, float addC, float addB) {
  float s = level[(i * NV1 + j) * NV1 + k];
  if (zs != 0.f) {
    bool ctr = (i == 49) && (j == 49) && (k == 49);
    bool bnd = (i <= 1) || (i >= RES - 1) || (j <= 1) || (j >= RES - 1) ||
               (k <= 1) || (k >= RES - 1);
    s += ctr ? addC : (bnd ? addB : 0.f);
  }
  return s;
}

// ---------------- kernel 1: global sdf statistics (block partials) ----------------
__global__ __launch_bounds__(256) void k_sdf_stats(const float* __restrict__ level,
                                                   float* __restrict__ part) {
  int tid = threadIdx.x;
  int gid = blockIdx.x * blockDim.x + tid;
  int stride = gridDim.x * blockDim.x;
  float mn = FLT_BIG, mx = -FLT_BIG, sa = 0.f, pos = 0.f, neg = 0.f;
  for (int v = gid; v < NVTOT; v += stride) {
    float s = level[v];
    mn = fminf(mn, s);
    mx = fmaxf(mx, s);
    sa += fabsf(s);
    int i = v / (NV1 * NV1);
    int r = v - i * (NV1 * NV1);
    int j = r / NV1;
    int k = r - j * NV1;
    if (i >= 1 && i <= RES - 1 && j >= 1 && j <= RES - 1 && k >= 1 && k <= RES - 1) {
      pos += (s > 0.f) ? 1.f : 0.f;
      neg += (s < 0.f) ? 1.f : 0.f;
    }
  }
  __shared__ float smn[256], smx[256], ssa[256], sp[256], sn[256];
  smn[tid] = mn; smx[tid] = mx; ssa[tid] = sa; sp[tid] = pos; sn[tid] = neg;
  __syncthreads();
  for (int s = 128; s > 0; s >>= 1) {
    if (tid < s) {
      smn[tid] = fminf(smn[tid], smn[tid + s]);
      smx[tid] = fmaxf(smx[tid], smx[tid + s]);
      ssa[tid] += ssa[tid + s];
      sp[tid]  += sp[tid + s];
      sn[tid]  += sn[tid + s];
    }
    __syncthreads();
  }
  if (tid == 0) {
    part[blockIdx.x * 5 + 0] = smn[0];
    part[blockIdx.x * 5 + 1] = smx[0];
    part[blockIdx.x * 5 + 2] = ssa[0];
    part[blockIdx.x * 5 + 3] = sp[0];
    part[blockIdx.x * 5 + 4] = sn[0];
  }
}

// ---------------- kernel 2: finalize scalars ----------------
__global__ __launch_bounds__(256) void k_sdf_finalize(const float* __restrict__ part,
                                                      float* __restrict__ scal,
                                                      float* __restrict__ out) {
  int tid = threadIdx.x;
  float mn = FLT_BIG, mx = -FLT_BIG, sa = 0.f, pos = 0.f, neg = 0.f;
  for (int p = tid; p < G1; p += 256) {
    mn = fminf(mn, part[p * 5 + 0]);
    mx = fmaxf(mx, part[p * 5 + 1]);
    sa += part[p * 5 + 2];
    pos += part[p * 5 + 3];
    neg += part[p * 5 + 4];
  }
  __shared__ float smn[256], smx[256], ssa[256], sp[256], sn[256];
  smn[tid] = mn; smx[tid] = mx; ssa[tid] = sa; sp[tid] = pos; sn[tid] = neg;
  __syncthreads();
  for (int s = 128; s > 0; s >>= 1) {
    if (tid < s) {
      smn[tid] = fminf(smn[tid], smn[tid + s]);
      smx[tid] = fmaxf(smx[tid], smx[tid + s]);
      ssa[tid] += ssa[tid + s];
      sp[tid]  += sp[tid + s];
      sn[tid]  += sn[tid + s];
    }
    __syncthreads();
  }
  if (tid == 0) {
    float minv = smn[0], maxv = smx[0];
    bool zs = (sp[0] == 0.f) || (sn[0] == 0.f);
    float empty = zs ? (ssa[0] / (float)NVTOT) : 0.f;
    scal[0] = minv;
    scal[1] = maxv;
    scal[2] = zs ? 1.f : 0.f;
    scal[3] = empty;
    scal[4] = 1.f - minv;    // center add
    scal[5] = -1.f - maxv;   // boundary add
    out[OFF_EMPTY] = empty;
    out[OFF_ZS]    = zs ? 1.f : 0.f;
  }
}

// ---------------- kernel 3: per-cube QEF ----------------
__global__ __launch_bounds__(256) void k_cubes(const float* __restrict__ level,
                                               const float* __restrict__ defo,
                                               const float* __restrict__ wn,
                                               const float* __restrict__ scal,
                                               float* __restrict__ out,
                                               float* __restrict__ part3) {
  const int tid = threadIdx.x;
  const int n = blockIdx.x * 256 + tid;   // NCUBE == gridDim*256 exactly

  // ---- stage this block's 256x21 weight rows into LDS (coalesced, async) ----
  __shared__ float wlds[256 * 21];
  const float* wsrc = wn + (size_t)blockIdx.x * (256 * 21);
#pragma unroll
  for (int it = 0; it < 21; ++it) {
    int idx = it * 256 + tid;
    async_cp_f32(wsrc + idx, &wlds[idx]);
  }
  async_wait_all();
  __syncthreads();

  const float zs   = scal[2];
  const float addC = scal[4];
  const float addB = scal[5];

  // cube coords
  const int ci = n / (RES * RES);
  const int rr = n - ci * (RES * RES);
  const int cj = rr / RES;
  const int ck = rr - cj * RES;

  // 8 corners: sdf and deformed positions
  float s8[8];
  float xx[8], xy[8], xz[8];
#pragma unroll
  for (int c = 0; c < 8; ++c) {
    int dx = c & 1, dy = (c >> 1) & 1, dz = (c >> 2) & 1;
    int i = ci + dx, j = cj + dy, k = ck + dz;
    s8[c] = sdf_at(level, i, j, k, zs, addC, addB);
    int vi = (i * NV1 + j) * NV1 + k;
    xx[c] = (float)i / (float)RES - 0.5f + DSCALE * tanhf(defo[3 * vi + 0]);
    xy[c] = (float)j / (float)RES - 0.5f + DSCALE * tanhf(defo[3 * vi + 1]);
    xz[c] = (float)k / (float)RES - 0.5f + DSCALE * tanhf(defo[3 * vi + 2]);
  }

  // weights
  const float* wr = &wlds[tid * 21];
  float beta[12], alpha[8];
#pragma unroll
  for (int e = 0; e < 12; ++e) beta[e] = 1.0f + WSCALE * tanhf(wr[e]);
#pragma unroll
  for (int c = 0; c < 8; ++c) alpha[c] = 1.0f + WSCALE * tanhf(wr[12 + c]);
  const float gamma = 1.0f + WSCALE * tanhf(wr[20]);

  const int EA[12] = {0, 1, 3, 2, 4, 5, 7, 6, 0, 1, 3, 2};
  const int EB[12] = {1, 3, 2, 0, 5, 7, 6, 4, 4, 5, 7, 6};

  float pex[12], pey[12], pez[12], crv[12];
  float wsum = 0.f, ax = 0.f, ay = 0.f, az = 0.f, crossSum = 0.f;
#pragma unroll
  for (int e = 0; e < 12; ++e) {
    int a = EA[e], b = EB[e];
    float sa = s8[a], sb = s8[b];
    float cross = ((sa < 0.f) != (sb < 0.f)) ? 1.f : 0.f;
    float wa = alpha[a] * fabsf(sb);
    float wb = alpha[b] * fabsf(sa);
    float den = wa + wb;
    den = (den == 0.f) ? 1.f : den;
    float t = wa / den;
    pex[e] = xx[a] * t + xx[b] * (1.f - t);
    pey[e] = xy[a] * t + xy[b] * (1.f - t);
    pez[e] = xz[a] * t + xz[b] * (1.f - t);
    crv[e] = cross;
    float w = beta[e] * cross;
    wsum += w;
    ax += w * pex[e];
    ay += w * pey[e];
    az += w * pez[e];
    crossSum += cross;
  }

  // centroid
  float cx = 0.f, cy = 0.f, cz = 0.f;
#pragma unroll
  for (int c = 0; c < 8; ++c) { cx += xx[c]; cy += xy[c]; cz += xz[c]; }
  cx *= 0.125f; cy *= 0.125f; cz *= 0.125f;

  float inv = 1.0f / (wsum + QEF_REG);
  float vdx = (ax + QEF_REG * cx) * inv;
  float vdy = (ay + QEF_REG * cy) * inv;
  float vdz = (az + QEF_REG * cz) * inv;

  // ---- stage vd through LDS for fully coalesced global stores ----
  __shared__ float vdl[256 * 3];
  vdl[3 * tid + 0] = vdx;
  vdl[3 * tid + 1] = vdy;
  vdl[3 * tid + 2] = vdz;

  // deviation / reg loss contribution
  float dev = 0.f;
#pragma unroll
  for (int e = 0; e < 12; ++e) {
    float dxp = pex[e] - vdx, dyp = pey[e] - vdy, dzp = pez[e] - vdz;
    dev += (dxp * dxp + dyp * dyp + dzp * dzp) * crv[e];
  }
  float surf = (crossSum > 0.f) ? 1.f : 0.f;
  float contrib = gamma * dev * surf;

  __shared__ float sdev[256], ssurf[256];
  sdev[tid] = contrib;
  ssurf[tid] = surf;
  __syncthreads();

  // coalesced vd store: 768 consecutive floats per block
  {
    size_t base = (size_t)OFF_VD + (size_t)blockIdx.x * 768;
#pragma unroll
    for (int it = 0; it < 3; ++it) {
      int i = it * 256 + tid;
      out[base + i] = vdl[i];
    }
  }

  for (int s = 128; s > 0; s >>= 1) {
    if (tid < s) { sdev[tid] += sdev[tid + s]; ssurf[tid] += ssurf[tid + s]; }
    __syncthreads();
  }
  if (tid == 0) {
    part3[blockIdx.x * 2 + 0] = sdev[0];
    part3[blockIdx.x * 2 + 1] = ssurf[0];
  }
}

// ---------------- kernel 4: reg_loss finalize ----------------
__global__ __launch_bounds__(256) void k_reg_loss(const float* __restrict__ part3,
                                                  float* __restrict__ out) {
  int tid = threadIdx.x;
  float sd = 0.f, ss = 0.f;
  for (int p = tid; p < G3; p += 256) {
    sd += part3[p * 2 + 0];
    ss += part3[p * 2 + 1];
  }
  __shared__ float a[256], b[256];
  a[tid] = sd; b[tid] = ss;
  __syncthreads();
  for (int s = 128; s > 0; s >>= 1) {
    if (tid < s) { a[tid] += a[tid + s]; b[tid] += b[tid + s]; }
    __syncthreads();
  }
  if (tid == 0) out[OFF_REG] = a[0] / fmaxf(b[0], 1.0f);
}

// ---------------- kernel 5: edges -> triangles + face mask ----------------
__global__ __launch_bounds__(256) void k_edges(const float* __restrict__ level,
                                               const float* __restrict__ scal,
                                               float* __restrict__ out) {
  const int tid = threadIdx.x;
  const int e0 = blockIdx.x * 256;
  const int e = e0 + tid;
  const int nvalid = (NEDGE - e0 < 256) ? (NEDGE - e0) : 256;

  __shared__ float tlds[256 * 6];
  __shared__ float mlds[256 * 2];

  if (e < NEDGE) {
    const float zs   = scal[2];
    const float addC = scal[4];
    const float addB = scal[5];

    int i0, j0, k0, i1, j1, k1;
    int q0, q1, q2, q3;
    if (e < NEF) {                       // X family: I full, J,K inter
      int f = e;
      int i = f / 9025;                  // 95*95
      int r = f - i * 9025;
      int j = r / 95 + 1;
      int k = r % 95 + 1;
      i0 = i; j0 = j; k0 = k; i1 = i + 1; j1 = j; k1 = k;
      q0 = (i * RES + (j - 1)) * RES + (k - 1);
      q1 = (i * RES + j) * RES + (k - 1);
      q2 = (i * RES + j) * RES + k;
      q3 = (i * RES + (j - 1)) * RES + k;
    } else if (e < 2 * NEF) {            // Y family: I inter, J full, K inter
      int f = e - NEF;
      int i = f / 9120 + 1;              // 96*95
      int r = f % 9120;
      int j = r / 95;
      int k = r % 95 + 1;
      i0 = i; j0 = j; k0 = k; i1 = i; j1 = j + 1; k1 = k;
      q0 = ((i - 1) * RES + j) * RES + (k - 1);
      q1 = ((i - 1) * RES + j) * RES + k;
      q2 = (i * RES + j) * RES + k;
      q3 = (i * RES + j) * RES + (k - 1);
    } else {                             // Z family: I,J inter, K full
      int f = e - 2 * NEF;
      int i = f / 9120 + 1;              // 95*96
      int r = f % 9120;
      int j = r / 96 + 1;
      int k = r % 96;
      i0 = i; j0 = j; k0 = k; i1 = i; j1 = j; k1 = k + 1;
      q0 = ((i - 1) * RES + (j - 1)) * RES + k;
      q1 = (i * RES + (j - 1)) * RES + k;
      q2 = (i * RES + j) * RES + k;
      q3 = ((i - 1) * RES + j) * RES + k;
    }

    float sA = sdf_at(level, i0, j0, k0, zs, addC, addB);
    float sB = sdf_at(level, i1, j1, k1, zs, addC, addB);
    bool emask = (sA < 0.f) != (sB < 0.f);
    bool flip = (sA < 0.f);
    int c1f = flip ? q3 : q1;
    int c3f = flip ? q1 : q3;

    float m = emask ? 1.f : 0.f;
    tlds[6 * tid + 0] = m * (float)q0;
    tlds[6 * tid + 1] = m * (float)c1f;
    tlds[6 * tid + 2] = m * (float)q2;
    tlds[6 * tid + 3] = m * (float)q0;
    tlds[6 * tid + 4] = m * (float)q2;
    tlds[6 * tid + 5] = m * (float)c3f;
    mlds[2 * tid + 0] = m;
    mlds[2 * tid + 1] = m;
  }
  __syncthreads();

  // coalesced block stores
  {
    size_t tb = (size_t)OFF_TRI + (size_t)6 * e0;
    int lim = 6 * nvalid;
#pragma unroll
    for (int it = 0; it < 6; ++it) {
      int i = it * 256 + tid;
      if (i < lim) out[tb + i] = tlds[i];
    }
    size_t mb = (size_t)OFF_MASK + (size_t)2 * e0;
    int lim2 = 2 * nvalid;
#pragma unroll
    for (int it = 0; it < 2; ++it) {
      int i = it * 256 + tid;
      if (i < lim2) out[mb + i] = mlds[i];
    }
  }
}

// ---------------- launcher ----------------
extern "C" void kernel_launch(void* const* d_in, const int* in_sizes, int n_in,
                              void* d_out, int out_size, void* d_ws, size_t ws_size,
                              hipStream_t stream) {
  const float* level = (const float*)d_in[0];
  const float* defo  = (const float*)d_in[1];
  const float* wn    = (const float*)d_in[2];
  float* out = (float*)d_out;
  float* ws  = (float*)d_ws;

  float* scal  = ws + WS_SCAL;
  float* part1 = ws + WS_PART1;
  float* part3 = ws + WS_PART3;

  k_sdf_stats<<<G1, 256, 0, stream>>>(level, part1);
  k_sdf_finalize<<<1, 256, 0, stream>>>(part1, scal, out);
  k_cubes<<<G3, 256, 0, stream>>>(level, defo, wn, scal, out, part3);
  k_reg_loss<<<1, 256, 0, stream>>>(part3, out);
  k_edges<<<(NEDGE + 255) / 256, 256, 0, stream>>>(level, scal, out);
}